// General_detector_57526791962855
// MI455X (gfx1250) — compile-verified
//
#include <hip/hip_runtime.h>
#include <cstdint>
#include <cstddef>

// ---------------- problem constants (from reference) ----------------
#define IMG_F      640.0f
#define CONF_THRE  0.05f
#define NMS_THRE   0.5f
#define MAX_DET    100
#define TOPKN      1000
#define NEGV       (-1000000000.0f)
#define CLS_OFF    4096.0f
#define NB         16
#define NA         25200
#define NC         80
#define BN         (NB * NA)

// ---------------- kernel A tiling ----------------
#define AROWS  128   // anchors per block (4 waves of 32)
#define ROWDW  85    // LDS row stride in dwords: 80 data + 5 pad dwords
                     // (TDM pad: 1 dword per 16 dwords; gcd(85,64)=1 -> conflict-free)

typedef unsigned int v4u __attribute__((ext_vector_type(4)));
typedef int          v8i __attribute__((ext_vector_type(8)));
typedef int          v4i __attribute__((ext_vector_type(4)));

// ====================================================================
// Kernel A: TDM-stage class scores to LDS, class max/argmax, box decode
// ====================================================================
__global__ __launch_bounds__(AROWS) void score_decode_kernel(
    const float* __restrict__ regs, const float* __restrict__ clses,
    const float* __restrict__ anchors, float* __restrict__ sc,
    float4* __restrict__ box, float4* __restrict__ aux)
{
  __shared__ float tile[AROWS * ROWDW];   // 43,520 B
  const int t = threadIdx.x;
  const int blockStart = blockIdx.x * AROWS;

  // ---- Tensor Data Mover: DMA AROWSx80 f32 tile (global -> LDS) ----
  if ((t >> 5) == 0) {  // wave 0 only (uniform branch)
    unsigned ldsAddr = (unsigned)(uintptr_t)(void*)tile;   // low 32 bits of flat LDS addr = LDS offset
    unsigned long long ga =
        (unsigned long long)(uintptr_t)(clses + (size_t)blockStart * NC);
    v4u g0;
    g0[0] = 1u;                                   // count=1 valid descriptor
    g0[1] = ldsAddr;                              // lds_addr (bytes)
    g0[2] = (unsigned)(ga & 0xFFFFFFFFu);         // global_addr[31:0]
    g0[3] = (unsigned)((ga >> 32) & 0x1FFFFFFu)   // global_addr[56:32]
          | (2u << 30);                           // type = 2 ("image")
    v8i g1;
    g1[0] = (2 << 16)      // data_size = 4 bytes
          | (1 << 20)      // pad_enable
          | (3 << 22);     // pad_interval code 3 = 16 dwords ; pad_amount code 0 = 1 dword
    g1[1] = (NC & 0xFFFF) << 16;      // tensor_dim0 lo16 (=80)
    g1[2] = (AROWS & 0xFFFF) << 16;   // tensor_dim0 hi16 (=0) | tensor_dim1 lo16 (=128)
    g1[3] = (NC << 16);               // tensor_dim1 hi16 (=0) | tile_dim0 (=80)
    g1[4] = AROWS;                    // tile_dim1 (=128) | tile_dim2 (=0)
    g1[5] = NC;                       // tensor_dim0_stride lo32 (=80 elements)
    g1[6] = 0;                        // stride hi | tensor_dim1_stride lo (unused, 2D)
    g1[7] = 0;
    v4i g2 = {0, 0, 0, 0};            // dims 2/3 unused (2D tensor)
    v4i g3 = {0, 0, 0, 0};
    v8i g4 = {0, 0, 0, 0, 0, 0, 0, 0};  // extra group in 6-arg builtin form
    __builtin_amdgcn_tensor_load_to_lds(g0, g1, g2, g3, g4, 0);
    __builtin_amdgcn_s_wait_tensorcnt(0);
  }
  __syncthreads();   // tile visible to all 4 waves

  const int gid = blockStart + t;       // flat (batch*NA + anchor)
  const int n   = gid % NA;             // anchor index within image

  // ---- ordered first-max argmax over 80 classes (matches jnp.argmax) ----
  const int rowBase = t * ROWDW;
  float best = -3.4e38f; int bi = 0;
#pragma unroll 8
  for (int j = 0; j < NC; ++j) {
    float v = tile[rowBase + j + (j >> 4)];   // +1 pad dword every 16 dwords
    if (v > best) { best = v; bi = j; }
  }

  // ---- box decode ----
  const float* r = regs + (size_t)gid * 5;
  float d0 = r[0], d1 = r[1], d2 = r[2], d3 = r[3], obj = r[4];
  const float4 a = *(const float4*)(anchors + (size_t)n * 4);
  float aw  = a.z - a.x, ah = a.w - a.y;
  float acx = (a.x + a.z) * 0.5f, acy = (a.y + a.w) * 0.5f;
  float cx  = acx + d0 * aw, cy = acy + d1 * ah;
  float w   = aw * expf(d2), h = ah * expf(d3);
  float x1 = fminf(fmaxf(cx - 0.5f * w, 0.f), IMG_F);
  float y1 = fminf(fmaxf(cy - 0.5f * h, 0.f), IMG_F);
  float x2 = fminf(fmaxf(cx + 0.5f * w, 0.f), IMG_F);
  float y2 = fminf(fmaxf(cy + 0.5f * h, 0.f), IMG_F);

  float s = obj * best;
  s = (s >= CONF_THRE) ? s : NEGV;

  sc[gid]  = s;
  box[gid] = make_float4(x1, y1, x2, y2);
  aux[gid] = make_float4(obj, best, (float)bi, 0.f);
}

// ====================================================================
// Kernel B: exact 32-bit radix-select top-1000 + deterministic gather
// ====================================================================
__device__ __forceinline__ unsigned fkey(float f) {
  unsigned u = __float_as_uint(f);
  unsigned mask = (unsigned)((int)u >> 31) | 0x80000000u;
  return u ^ mask;   // monotonic: larger float -> larger unsigned
}

__global__ __launch_bounds__(1024) void topk_kernel(
    const float* __restrict__ sc, int* __restrict__ topk)
{
  const int b   = blockIdx.x;
  const int tid = threadIdx.x;
  const float* s = sc + (size_t)b * NA;

  __shared__ int      hist[256];
  __shared__ unsigned scan[1024];
  __shared__ unsigned sPrefix;
  __shared__ int      sAbove;

  if (tid == 0) { sPrefix = 0; sAbove = 0; }

  // ---- 4 byte-rounds of radix select (exact 32-bit cutoff) ----
  for (int p = 0; p < 4; ++p) {
    if (tid < 256) hist[tid] = 0;
    __syncthreads();
    unsigned pref = sPrefix;
    for (int i = tid; i < NA; i += 1024) {
      unsigned k = fkey(s[i]);
      bool match = (p == 0) || ((k >> (32 - 8 * p)) == pref);
      if (match) atomicAdd(&hist[(k >> (24 - 8 * p)) & 0xFF], 1);
    }
    __syncthreads();
    if (tid == 0) {
      int K = TOPKN - sAbove;   // still needed within matching set
      int cum = 0, bin = 255;
      for (; bin >= 0; --bin) {
        int c = hist[bin];
        if (cum + c >= K) break;
        cum += c;
      }
      sAbove += cum;
      sPrefix = (sPrefix << 8) | (unsigned)bin;
    }
    __syncthreads();
  }
  const unsigned V    = sPrefix;         // exact cutoff key
  const int above    = sAbove;           // # keys strictly > V
  const int need     = TOPKN - above;    // # keys == V to take (index order)

  // ---- deterministic index-ordered gather (block scans, no global atomics) ----
  int baseA = 0, baseB = 0;
  for (int c0 = 0; c0 < NA; c0 += 1024) {
    int i = c0 + tid;
    unsigned k = (i < NA) ? fkey(s[i]) : 0u;
    bool fa = (i < NA) && (k > V);
    bool fb = (i < NA) && (k == V);
    unsigned v = (fa ? 1u : 0u) | (fb ? 0x10000u : 0u);
    scan[tid] = v;
    __syncthreads();
    for (int off = 1; off < 1024; off <<= 1) {
      unsigned add = (tid >= off) ? scan[tid - off] : 0u;
      __syncthreads();
      scan[tid] += add;
      __syncthreads();
    }
    unsigned incl = scan[tid];
    unsigned tot  = scan[1023];
    unsigned excl = incl - v;
    if (fa) {
      topk[(size_t)b * TOPKN + baseA + (int)(excl & 0xFFFF)] = i;
    } else if (fb) {
      int pb = baseB + (int)(excl >> 16);
      if (pb < need) topk[(size_t)b * TOPKN + above + pb] = i;
    }
    baseA += (int)(tot & 0xFFFF);
    baseB += (int)(tot >> 16);
    __syncthreads();
  }
}

// ====================================================================
// Kernel C: greedy NMS (100 rounds) + output assembly
// ====================================================================
__global__ __launch_bounds__(1024) void nms_kernel(
    const float* __restrict__ sc, const float4* __restrict__ box,
    const float4* __restrict__ aux, const int* __restrict__ topk,
    float* __restrict__ out)
{
  const int b   = blockIdx.x;
  const int tid = threadIdx.x;

  __shared__ float ox1[TOPKN], oy1[TOPKN], ox2[TOPKN], oy2[TOPKN];
  __shared__ float rs[1024];
  __shared__ int   ri[1024];

  float myScore = NEGV;
  float bx0 = 0, bx1 = 0, bx2 = 0, bx3 = 0, obj = 0, cc = 0, cp = 0, myArea = 0;
  if (tid < TOPKN) {
    int idx = topk[(size_t)b * TOPKN + tid];
    size_t g = (size_t)b * NA + idx;
    myScore = sc[g];
    float4 bb = box[g];
    float4 ax = aux[g];
    obj = ax.x; cc = ax.y; cp = ax.z;
    bx0 = bb.x; bx1 = bb.y; bx2 = bb.z; bx3 = bb.w;
    float off = cp * CLS_OFF;           // class-offset boxes for IoU
    float X1 = bb.x + off, Y1 = bb.y + off, X2 = bb.z + off, Y2 = bb.w + off;
    ox1[tid] = X1; oy1[tid] = Y1; ox2[tid] = X2; oy2[tid] = Y2;
    myArea = (X2 - X1) * (Y2 - Y1);
  }
  __syncthreads();

  for (int it = 0; it < MAX_DET; ++it) {
    rs[tid] = myScore; ri[tid] = tid;
    __syncthreads();
    for (int off = 512; off > 0; off >>= 1) {
      if (tid < off) {
        float o = rs[tid + off]; int oi = ri[tid + off];
        float m = rs[tid];       int mi = ri[tid];
        if (o > m || (o == m && oi < mi)) { rs[tid] = o; ri[tid] = oi; }
      }
      __syncthreads();
    }
    int   best   = ri[0];
    float bscore = rs[0];
    __syncthreads();   // all reads of rs/ri done before next-iteration writes

    bool valid = bscore > NEGV * 0.5f;
    float B1 = ox1[best], B2 = oy1[best], B3 = ox2[best], B4 = oy2[best];

    if (tid == best) {
      float* o = out + ((size_t)b * MAX_DET + it) * 7;
      if (valid) {
        o[0] = bx0; o[1] = bx1; o[2] = bx2; o[3] = bx3;
        o[4] = obj; o[5] = cc;  o[6] = cp;
      } else {
        o[0] = 0; o[1] = 0; o[2] = 0; o[3] = 0; o[4] = 0; o[5] = 0; o[6] = 0;
      }
    }
    if (tid < TOPKN) {
      float lx = fmaxf(B1, ox1[tid]);
      float ly = fmaxf(B2, oy1[tid]);
      float rx = fminf(B3, ox2[tid]);
      float ry = fminf(B4, oy2[tid]);
      float iw = fmaxf(rx - lx, 0.f);
      float ih = fmaxf(ry - ly, 0.f);
      float inter = iw * ih;
      float areaB = (B3 - B1) * (B4 - B2);
      float uni   = areaB + myArea - inter;
      float iou   = (uni > 0.f) ? (inter / uni) : 0.f;
      if (iou > NMS_THRE) myScore = NEGV;
      if (tid == best)    myScore = NEGV;
    }
  }
}

// ====================================================================
extern "C" void kernel_launch(void* const* d_in, const int* in_sizes, int n_in,
                              void* d_out, int out_size, void* d_ws, size_t ws_size,
                              hipStream_t stream)
{
  const float* regs    = (const float*)d_in[0];   // (16, 25200, 5)
  const float* clses   = (const float*)d_in[1];   // (16, 25200, 80)
  const float* anchors = (const float*)d_in[2];   // (25200, 4)
  float* out = (float*)d_out;                     // (16, 100, 7)

  // workspace layout (16B-aligned sections)
  float*  sc   = (float*)d_ws;                               // BN floats
  float4* box  = (float4*)((char*)d_ws + (size_t)BN * 4);    // BN float4
  float4* aux  = box + BN;                                   // BN float4
  int*    topk = (int*)(aux + BN);                           // NB*1000 ints

  score_decode_kernel<<<BN / AROWS, AROWS, 0, stream>>>(regs, clses, anchors,
                                                        sc, box, aux);
  topk_kernel<<<NB, 1024, 0, stream>>>(sc, topk);
  nms_kernel<<<NB, 1024, 0, stream>>>(sc, box, aux, topk, out);
}